// SRULayer_4612794876183
// MI455X (gfx1250) — compile-verified
//
#include <hip/hip_runtime.h>
#include <cstdint>

#define T_DIM 1024
#define B_DIM 8
#define D_DIM 2048
#define M_DIM (T_DIM * B_DIM)   // 8192 rows of x
#define N_DIM (3 * D_DIM)       // 6144 output features
#define K_DIM D_DIM             // 2048 reduction

#define BM 64
#define BN 128
#define BK 32
#define LDSTR 40                // 32 data floats + 8 pad floats -> 160B rows (16B aligned)

typedef __attribute__((ext_vector_type(2))) float v2f;
typedef __attribute__((ext_vector_type(4))) float v4f;
typedef __attribute__((ext_vector_type(8))) float v8f;
typedef uint32_t u32;
typedef uint64_t u64;

#if defined(__has_builtin)
#if __has_builtin(__builtin_amdgcn_tensor_load_to_lds) && \
    __has_builtin(__builtin_amdgcn_s_wait_tensorcnt)
#define USE_TDM 1
#endif
#endif

#ifdef USE_TDM
typedef __attribute__((ext_vector_type(4))) u32 u32x4;
typedef __attribute__((ext_vector_type(8))) int i32x8;
typedef __attribute__((ext_vector_type(4))) int i32x4;

// Issue one TDM 2D tile load: tile is (BK x tile_rows) fp32, row stride
// `row_stride_elems` in global, written to LDS with an 8-DWORD pad after
// every 32 DWORDs (-> LDSTR=40 float rows). D# bitfields per CDNA5 ISA §8.3/8.4.
// clang-23 (amdgpu-toolchain) 6-arg form: (g0, g1, g2, g3, <i32x8>, cpol).
__device__ __forceinline__ void tdm_load_tile(u32 lds_byte_off, const float* gptr,
                                              u32 tile_rows, u32 row_stride_elems) {
  const u64 ga = (u64)(uintptr_t)gptr;
  u32x4 g0;
  g0.x = 1u;                                            // count=1, user descriptor
  g0.y = lds_byte_off;                                  // lds_addr [63:32]
  g0.z = (u32)ga;                                       // global_addr [95:64]
  g0.w = ((u32)(ga >> 32) & 0x01FFFFFFu) | (2u << 30);  // addr[56:32] | type=2

  const u32 dim1 = 1u << 20;                            // large: no OOB in-tile
  i32x8 g1;
  g1[0] = (int)((2u << 16)        // data_size = 4 bytes
              | (1u << 20)        // pad_enable
              | (4u << 22)        // pad_interval: 1<<4 8B-units = 32 DWORDs
              | (7u << 25));      // pad_amount: 8 DWORDs
  g1[1] = (int)(((u32)K_DIM & 0xFFFFu) << 16);                    // tensor_dim0 lo16
  g1[2] = (int)(((u32)K_DIM >> 16) | ((dim1 & 0xFFFFu) << 16));   // dim0 hi | dim1 lo
  g1[3] = (int)((dim1 >> 16) | ((u32)BK << 16));                  // dim1 hi | tile_dim0
  g1[4] = (int)(tile_rows);                                       // tile_dim1 | tile_dim2=0
  g1[5] = (int)row_stride_elems;                                  // tensor_dim0_stride lo32
  g1[6] = 0;
  g1[7] = 0;
  const i32x4 z4 = {0, 0, 0, 0};
  const i32x8 z8 = {0, 0, 0, 0, 0, 0, 0, 0};
  __builtin_amdgcn_tensor_load_to_lds(g0, g1, z4, z4, z8, 0);
}
#endif

// ---------------------------------------------------------------------------
// Kernel 1: ufr[m, n] = sum_k x[m, k] * W[n, k]   (fp32, exact, WMMA 16x16x4)
// ---------------------------------------------------------------------------
__global__ __launch_bounds__(256) void sru_gemm(const float* __restrict__ A,
                                                const float* __restrict__ Bw,
                                                float* __restrict__ C) {
  __shared__ float As[BM][LDSTR];
  __shared__ float Bs[BN][LDSTR];

  const int tid  = threadIdx.x;
  const int lane = tid & 31;
  const int wave = tid >> 5;          // 8 waves
  const int half = lane >> 4;         // 0: lanes 0-15, 1: lanes 16-31
  const int lrow = lane & 15;
  const int wm   = wave & 3;          // 4 waves along M
  const int wn   = wave >> 2;         // 2 waves along N
  const int m0   = blockIdx.y * BM;
  const int n0   = blockIdx.x * BN;

  v8f acc[4] = {};                    // 16x64 per-wave output strip

  for (int kk = 0; kk < K_DIM; kk += BK) {
#ifdef USE_TDM
    if (wave == 0) {
      tdm_load_tile((u32)(uintptr_t)&As[0][0], A  + (size_t)m0 * K_DIM + kk, BM, K_DIM);
      tdm_load_tile((u32)(uintptr_t)&Bs[0][0], Bw + (size_t)n0 * K_DIM + kk, BN, K_DIM);
      __builtin_amdgcn_s_wait_tensorcnt(0);
    }
    __syncthreads();
#else
    #pragma unroll
    for (int j = tid; j < (BM * BK) / 4; j += 256) {
      const int row = j >> 3, c4 = (j & 7) * 4;
      *(v4f*)&As[row][c4] = *(const v4f*)&A[(size_t)(m0 + row) * K_DIM + kk + c4];
    }
    #pragma unroll
    for (int j = tid; j < (BN * BK) / 4; j += 256) {
      const int row = j >> 3, c4 = (j & 7) * 4;
      *(v4f*)&Bs[row][c4] = *(const v4f*)&Bw[(size_t)(n0 + row) * K_DIM + kk + c4];
    }
    __syncthreads();
#endif

    const int arow = wm * 16 + lrow;
    #pragma unroll
    for (int k4 = 0; k4 < BK; k4 += 4) {
      const int kc = k4 + 2 * half;   // A VGPR0/1: K={0,1} (lo lanes) / K={2,3} (hi lanes)
      const v2f a = *(const v2f*)&As[arow][kc];
      #pragma unroll
      for (int nf = 0; nf < 4; ++nf) {
        const v2f b = *(const v2f*)&Bs[wn * 64 + nf * 16 + lrow][kc];
        acc[nf] = __builtin_amdgcn_wmma_f32_16x16x4_f32(
            /*neg_a=*/false, a, /*neg_b=*/false, b,
            /*c_mod=*/(short)0, acc[nf], /*reuse_a=*/false, /*reuse_b=*/false);
      }
    }
    __syncthreads();
  }

  // C/D layout: VGPR r holds M=r (lanes 0-15) and M=r+8 (lanes 16-31), N=lane&15
  const int mbase = m0 + wm * 16 + half * 8;
  #pragma unroll
  for (int nf = 0; nf < 4; ++nf) {
    const int n = n0 + wn * 64 + nf * 16 + lrow;
    #pragma unroll
    for (int r = 0; r < 8; ++r) {
      C[(size_t)(mbase + r) * N_DIM + n] = acc[nf][r];
    }
  }
}

// ---------------------------------------------------------------------------
// Kernel 2: sequential SRU scan over T; one thread per (b, d) column.
// Writes h into out[0 : T*B*D], final cell state into cT.
// ---------------------------------------------------------------------------
__global__ __launch_bounds__(256) void sru_scan(const float* __restrict__ x,
                                                const float* __restrict__ c0,
                                                const float* __restrict__ ufr,
                                                float* __restrict__ h_out,
                                                float* __restrict__ cT_out) {
  const int idx = blockIdx.x * blockDim.x + threadIdx.x;  // 0 .. B*D-1
  if (idx >= B_DIM * D_DIM) return;
  const int b = idx >> 11;          // / D_DIM
  const int d = idx & (D_DIM - 1);

  float c = c0[idx];
  const float* up = ufr + (size_t)b * N_DIM + d;  // u at +0, f at +D, r at +2D
  const float* xp = x + idx;
  float* hp = h_out + idx;

  for (int t = 0; t < T_DIM; ++t) {
    const float u  = up[0];
    const float fz = up[D_DIM];
    const float rz = up[2 * D_DIM];
    const float xv = xp[0];
    const float f = 1.0f / (1.0f + expf(-fz));
    const float r = 1.0f / (1.0f + expf(-rz));
    c = f * c + (1.0f - f) * u;
    hp[0] = r * tanhf(c) + (1.0f - r) * xv;
    up += (size_t)B_DIM * N_DIM;
    xp += (size_t)B_DIM * D_DIM;
    hp += (size_t)B_DIM * D_DIM;
  }
  cT_out[idx] = c;
}

// ---------------------------------------------------------------------------
// Kernel 3: exact-erf GELU + LayerNorm over D, in place on out. One block/row.
// ---------------------------------------------------------------------------
__global__ __launch_bounds__(256) void sru_gelu_ln(float* __restrict__ out,
                                                   const float* __restrict__ gamma,
                                                   const float* __restrict__ beta) {
  const int row = blockIdx.x;                 // t*B + b
  float* h = out + (size_t)row * D_DIM;
  const int tid = threadIdx.x;

  float g[D_DIM / 256];
  float s = 0.0f, s2 = 0.0f;
  #pragma unroll
  for (int j = 0; j < D_DIM / 256; ++j) {
    const int i = tid + j * 256;
    const float v = h[i];
    const float gg = 0.5f * v * (1.0f + erff(v * 0.70710678118654752440f));
    g[j] = gg;
    s += gg;
    s2 += gg * gg;
  }

  __shared__ float sb[256];
  __shared__ float sb2[256];
  sb[tid] = s;
  sb2[tid] = s2;
  __syncthreads();
  for (int off = 128; off > 0; off >>= 1) {
    if (tid < off) {
      sb[tid]  += sb[tid + off];
      sb2[tid] += sb2[tid + off];
    }
    __syncthreads();
  }
  const float mean = sb[0] * (1.0f / D_DIM);
  const float var  = sb2[0] * (1.0f / D_DIM) - mean * mean;
  const float inv  = rsqrtf(var + 1e-5f);

  #pragma unroll
  for (int j = 0; j < D_DIM / 256; ++j) {
    const int i = tid + j * 256;
    h[i] = (g[j] - mean) * inv * gamma[i] + beta[i];
  }
}

// ---------------------------------------------------------------------------
extern "C" void kernel_launch(void* const* d_in, const int* in_sizes, int n_in,
                              void* d_out, int out_size, void* d_ws, size_t ws_size,
                              hipStream_t stream) {
  const float* x     = (const float*)d_in[0];  // [T, B, D]
  const float* c0    = (const float*)d_in[1];  // [B, D]
  const float* W     = (const float*)d_in[2];  // [3D, D]
  const float* gamma = (const float*)d_in[3];  // [D]
  const float* beta  = (const float*)d_in[4];  // [D]

  float* out = (float*)d_out;                  // [T, B, D] then [B, D] (c_T)
  float* cT  = out + (size_t)T_DIM * B_DIM * D_DIM;
  float* ufr = (float*)d_ws;                   // [T*B, 3D] = 192 MB scratch

  dim3 ggrid(N_DIM / BN, M_DIM / BM);          // (48, 128)
  sru_gemm<<<ggrid, 256, 0, stream>>>(x, W, ufr);

  sru_scan<<<(B_DIM * D_DIM) / 256, 256, 0, stream>>>(x, c0, ufr, out, cT);

  sru_gelu_ln<<<T_DIM * B_DIM, 256, 0, stream>>>(out, gamma, beta);
}